// SmoothLoss_8821862826419
// MI455X (gfx1250) — compile-verified
//
#include <hip/hip_runtime.h>
#include <stdint.h>

// Problem constants (fixed by the reference setup)
#define BN 16
#define HH 512
#define WW 512
#define TH 32
#define TW 64
#define LH (TH + 2)          // +2 bottom halo rows (dy in [0,2])
#define LW (TW + 4)          // +-2 col halo (dx in [-2,2])
#define PLANE (LH * LW)
#define NCELL (3 * PLANE)
#define NTHREADS 256
#define NTX (WW / TW)        // 8
#define NTY (HH / TH)        // 16
#define NPART (BN * NTX * NTY)  // 2048 block partials

typedef float v2f __attribute__((ext_vector_type(2)));
typedef float v8f __attribute__((ext_vector_type(8)));

// Canonical half of the 24 offsets (each +- pair contributes identically;
// factor 2 folded into the mean weights below).
static constexpr int DYS[12] = {1, 0, 1, 1, 2, 0, 2, 2, 1, 1, 2, 2};
static constexpr int DXS[12] = {0, 1, 1, -1, 0, 2, 1, -1, 2, -2, 2, -2};

__global__ __launch_bounds__(NTHREADS)
void smooth_tile_kernel(const float* __restrict__ in,
                        const float* __restrict__ outp,
                        float* __restrict__ part) {
  __shared__ float smem[6 * PLANE];   // planes 0..2: ycc, planes 3..5: output
  __shared__ float red[NTHREADS];

  const int tid = threadIdx.x;
  const int b = blockIdx.z;
  const int h0 = blockIdx.y * TH;
  const int w0 = blockIdx.x * TW;

  // ---- Stage YCbCr planes (computed from flat-triple grouping) ----
  for (int idx = tid; idx < NCELL; idx += NTHREADS) {
    int c = idx / PLANE;
    int rem = idx - c * PLANE;
    int lh = rem / LW;
    int lw = rem - lh * LW;
    int gh = h0 + lh;
    int gw = w0 - 2 + lw;
    float val = 0.0f;
    if (gh < HH && (unsigned)gw < (unsigned)WW) {
      int i = ((b * 3 + c) * HH + gh) * WW + gw;
      int base = (i / 3) * 3;
      int j = i - base;  // 0,1,2 -> column of the 3x3 matrix
      float x0 = in[base], x1 = in[base + 1], x2 = in[base + 2];
      float m0 = (j == 0) ? 0.257f : ((j == 1) ? -0.148f : 0.439f);
      float m1 = (j == 0) ? 0.564f : ((j == 1) ? -0.291f : -0.368f);
      float m2 = (j == 0) ? 0.098f : ((j == 1) ? 0.439f : -0.071f);
      float bia = (j == 0) ? (16.0f / 255.0f) : (128.0f / 255.0f);
      val = fmaf(x0, m0, fmaf(x1, m1, fmaf(x2, m2, bia)));
    }
    smem[idx] = val;
  }

  // ---- Stage output planes via CDNA5 async global->LDS copies ----
  for (int idx = tid; idx < NCELL; idx += NTHREADS) {
    int c = idx / PLANE;
    int rem = idx - c * PLANE;
    int lh = rem / LW;
    int lw = rem - lh * LW;
    int gh = h0 + lh;
    int gw = w0 - 2 + lw;
    if (gh < HH && (unsigned)gw < (unsigned)WW) {
      int i = ((b * 3 + c) * HH + gh) * WW + gw;
      unsigned ldsb = (unsigned)(size_t)(&smem[3 * PLANE + idx]);
      asm volatile("global_load_async_to_lds_b32 %0, %1, %2"
                   :: "v"(ldsb), "v"(i * 4), "s"(outp)
                   : "memory");
    } else {
      smem[3 * PLANE + idx] = 0.0f;  // halo outside the image (never read)
    }
  }
  asm volatile("s_wait_asynccnt 0" ::: "memory");
  __syncthreads();

  // ---- Stencil: each thread owns one column x 8 consecutive rows ----
  const int tx = tid & 63;
  const int ty0 = (tid >> 6) * 8;
  const int gw = w0 + tx;
  const float SC = -1.0f / (2.0f * 10.0f * 10.0f);
  float p = 0.0f;

  for (int r = 0; r < 8; ++r) {
    int row = ty0 + r;
    int gh = h0 + row;
    const float* yC = &smem[row * LW + (tx + 2)];
    const float* oC = &smem[3 * PLANE + row * LW + (tx + 2)];
    float y0 = yC[0], y1 = yC[PLANE], y2 = yC[2 * PLANE];
    float o0 = oC[0], o1 = oC[PLANE], o2 = oC[2 * PLANE];
#pragma unroll
    for (int k = 0; k < 12; ++k) {
      const int dy = DYS[k], dx = DXS[k];
      if ((gh + dy) < HH && (unsigned)(gw + dx) < (unsigned)WW) {
        const int noff = dy * LW + dx;
        float d0 = yC[noff] - y0;
        float d1 = yC[noff + PLANE] - y1;
        float d2 = yC[noff + 2 * PLANE] - y2;
        float ss = fmaf(d0, d0, fmaf(d1, d1, d2 * d2));
        float wgt = __expf(ss * SC);
        float l1 = fabsf(oC[noff] - o0) + fabsf(oC[noff + PLANE] - o1) +
                   fabsf(oC[noff + 2 * PLANE] - o2);
        // factor 2 for the +- offset pair folded into the mean weight
        const int ady = dy < 0 ? -dy : dy;
        const int adx = dx < 0 ? -dx : dx;
        const float inv = 2.0f / (16.0f * (float)(HH - ady) * (float)(WW - adx));
        p = fmaf(wgt * l1, inv, p);
      }
    }
  }

  red[tid] = p;
  __syncthreads();

  // ---- Block reduction via chained V_WMMA_F32_16X16X4_F32 (B = ones) ----
  if (tid < 32) {
    v8f acc = {};
    v2f ones;
    ones.x = 1.0f;
    ones.y = 1.0f;
#pragma unroll
    for (int i = 0; i < NTHREADS / 64; ++i) {
      v2f a;
      a.x = red[i * 64 + tid];
      a.y = red[i * 64 + 32 + tid];
      acc = __builtin_amdgcn_wmma_f32_16x16x4_f32(false, a, false, ones,
                                                  (short)0, acc, false, false);
    }
    // total = sum over rows M of D[M][0]:
    // lane 0 holds M=0..7 (8 VGPRs), lane 16 holds M=8..15
    float s = acc[0] + acc[1] + acc[2] + acc[3] +
              acc[4] + acc[5] + acc[6] + acc[7];
    float s16 = __shfl(s, 16, 32);
    if (tid == 0) {
      int blk = (blockIdx.z * gridDim.y + blockIdx.y) * gridDim.x + blockIdx.x;
      part[blk] = s + s16;
    }
  }
}

__global__ __launch_bounds__(32)
void smooth_reduce_kernel(const float* __restrict__ part,
                          float* __restrict__ out) {
  const int tid = threadIdx.x;
  v8f acc = {};
  v2f ones;
  ones.x = 1.0f;
  ones.y = 1.0f;
  for (int i = 0; i < NPART / 64; ++i) {  // 32 chained WMMAs, fixed order
    v2f a;
    a.x = part[i * 64 + tid];
    a.y = part[i * 64 + 32 + tid];
    acc = __builtin_amdgcn_wmma_f32_16x16x4_f32(false, a, false, ones,
                                                (short)0, acc, false, false);
  }
  float s = acc[0] + acc[1] + acc[2] + acc[3] +
            acc[4] + acc[5] + acc[6] + acc[7];
  float s16 = __shfl(s, 16, 32);
  if (tid == 0) out[0] = s + s16;
}

extern "C" void kernel_launch(void* const* d_in, const int* in_sizes, int n_in,
                              void* d_out, int out_size, void* d_ws, size_t ws_size,
                              hipStream_t stream) {
  const float* in = (const float*)d_in[0];
  const float* outp = (const float*)d_in[1];
  float* part = (float*)d_ws;     // NPART floats = 8 KB of scratch
  float* res = (float*)d_out;     // single f32 scalar

  dim3 grid(NTX, NTY, BN);
  smooth_tile_kernel<<<grid, NTHREADS, 0, stream>>>(in, outp, part);
  smooth_reduce_kernel<<<1, 32, 0, stream>>>(part, res);
}